// GetCost_43069932044811
// MI455X (gfx1250) — compile-verified
//
#include <hip/hip_runtime.h>
#include <hip/hip_bf16.h>
#include <stdint.h>

// ---------------- configuration (matches reference constants) ----------------
#define BDIM    256          // threads per block = pixels per block
#define ROWS    256          // rows (pixels) staged per chunk
#define DPL     32           // D_PLANES
#define PAD     36           // padded row length in floats: 144B, 16B-aligned, low bank conflicts
#define NGRP    8            // G
#define MAXND   8            // compile-time cap on CostNum
#define HW_CONST (256 * 320) // H*W

// ---------------- gfx1250 async-copy + wait helpers --------------------------
#if defined(__has_builtin)
#if __has_builtin(__builtin_amdgcn_global_load_async_to_lds_b128)
#define ASYNC_B128_BUILTIN 1
#endif
#if __has_builtin(__builtin_amdgcn_s_wait_asynccnt)
#define WAIT_ASYNC_BUILTIN 1
#endif
#endif

typedef int v4i __attribute__((vector_size(16)));
typedef __attribute__((address_space(1))) v4i* g_v4i_ptr;  // global
typedef __attribute__((address_space(3))) v4i* l_v4i_ptr;  // LDS

__device__ __forceinline__ void async_copy16(const float* gsrc, uint32_t lds_byte_addr) {
#ifdef ASYNC_B128_BUILTIN
  __builtin_amdgcn_global_load_async_to_lds_b128(
      (g_v4i_ptr)(uintptr_t)gsrc,
      (l_v4i_ptr)(uintptr_t)lds_byte_addr,
      /*imm offset*/ 0, /*cpol*/ 0);
#else
  asm volatile("global_load_async_to_lds_b128 %0, %1, off"
               :: "v"(lds_byte_addr), "v"((uint64_t)(uintptr_t)gsrc)
               : "memory");
#endif
}

template <int N>
__device__ __forceinline__ void wait_asynccnt() {
#ifdef WAIT_ASYNC_BUILTIN
  __builtin_amdgcn_s_wait_asynccnt(N);
#else
  asm volatile("s_wait_asynccnt %0" :: "n"(N) : "memory");
#endif
}

// ---------------- kernel ------------------------------------------------------
__global__ __launch_bounds__(BDIM) void getcost_kernel(
    const float* __restrict__ dv,    // depth_values  (B,1,H,W)
    const float* __restrict__ pro0,  // (B*H*W, 8, 32)
    const float* __restrict__ pro1,  // (B*H*W, 1, 32)
    const float* __restrict__ di,    // depth_interval (B,1,H,W)
    const int*   __restrict__ cn,    // CostNum (1 element)
    float*       __restrict__ out)   // (B, 9*nd, H, W)
{
  __shared__ float smem[2 * ROWS * PAD];  // 73728 B, double-buffered chunk stage

  const int tid     = threadIdx.x;
  const int pixBase = blockIdx.x * ROWS;
  const int pix     = pixBase + tid;

  int nd = cn[0];
  nd = (nd < 2) ? 2 : ((nd > MAXND) ? MAXND : nd);

  constexpr float INV_MIN = 1.0f / 935.0f;  // 1/DEPTH_MAX
  constexpr float INV_MAX = 1.0f / 425.0f;  // 1/DEPTH_MIN
  const float SCALE = (float)(DPL - 1) / (INV_MAX - INV_MIN + 1e-10f);

  // ---- per-pixel sample indices / lerp weights (mirrors reference math) ----
  int   i0c[MAXND], i1c[MAXND];
  float w0v[MAXND], w1v[MAXND];
  {
    const float cur     = 1.0f / dv[pix];
    const float itv     = di[pix];
    const float low     = cur - (0.5f * (float)nd) * itv;
    const float new_itv = ((float)nd * itv) / (float)(nd - 1);
#pragma unroll
    for (int d = 0; d < MAXND; ++d) {
      const float samp_inv = low + (float)d * new_itv;
      const float depth    = 1.0f / samp_inv;             // reference round-trip
      const float x        = (1.0f / depth - INV_MIN) * SCALE;
      const float x0       = floorf(x);
      const float w1       = x - x0;
      const bool  v0       = (x0 >= 0.0f) && (x0 <= (float)(DPL - 1));
      const bool  v1       = ((x0 + 1.0f) >= 0.0f) && ((x0 + 1.0f) <= (float)(DPL - 1));
      int i0 = (int)x0;
      int i1 = i0 + 1;
      i0 = i0 < 0 ? 0 : (i0 > DPL - 1 ? DPL - 1 : i0);
      i1 = i1 < 0 ? 0 : (i1 > DPL - 1 ? DPL - 1 : i1);
      i0c[d] = i0;
      i1c[d] = i1;
      w0v[d] = v0 ? (1.0f - w1) : 0.0f;
      w1v[d] = v1 ? w1 : 0.0f;
    }
  }

  const int b  = pix / HW_CONST;
  const int hw = pix - b * HW_CONST;
  const int NC = NGRP + 1;  // 9 chunks: pro1 then pro0 g=0..7
  float* const outb = out + (size_t)b * (size_t)(NC * nd) * HW_CONST + hw;

  const uint32_t lds_base = (uint32_t)(uintptr_t)smem;

  // stage chunk c (256 rows x 32 floats) into buffer `sel` via async copies
  auto stage = [&](int c, int sel) {
    const float* src;
    int rs;  // row stride in floats
    if (c == 0) { src = pro1 + (size_t)pixBase * DPL;                              rs = DPL;        }
    else        { src = pro0 + (size_t)pixBase * (NGRP * DPL) + (size_t)(c - 1) * DPL; rs = NGRP * DPL; }
    const uint32_t bufb = lds_base + (uint32_t)sel * (uint32_t)(ROWS * PAD * 4);
#pragma unroll
    for (int i = 0; i < 8; ++i) {          // 2048 float4 slots / 256 threads
      const int flat = tid + i * BDIM;
      const int r    = flat >> 3;          // row within chunk
      const int q    = flat & 7;           // float4 within row
      async_copy16(src + (size_t)r * rs + q * 4,
                   bufb + (uint32_t)(r * PAD + q * 4) * 4u);
    }
  };

  stage(0, 0);
  for (int c = 0; c < NC; ++c) {
    if (c + 1 < NC) {
      stage(c + 1, (c + 1) & 1);   // prefetch next chunk into the other buffer
      wait_asynccnt<8>();          // loads complete in order: chunk c has landed
    } else {
      wait_asynccnt<0>();
    }
    __syncthreads();               // cross-wave: everyone's copies are visible

    const float* row = &smem[(c & 1) * (ROWS * PAD) + tid * PAD];
    float* oc = outb + (size_t)(c * nd) * HW_CONST;  // channel base = c*nd
#pragma unroll
    for (int d = 0; d < MAXND; ++d) {
      if (d < nd) {
        const float v = row[i0c[d]] * w0v[d] + row[i1c[d]] * w1v[d];
        oc[(size_t)d * HW_CONST] = v;
      }
    }
    __syncthreads();               // buffer (c&1) may be overwritten next iter
  }
}

// ---------------- launch ------------------------------------------------------
extern "C" void kernel_launch(void* const* d_in, const int* in_sizes, int n_in,
                              void* d_out, int out_size, void* d_ws, size_t ws_size,
                              hipStream_t stream) {
  const float* dv   = (const float*)d_in[0];  // depth_values
  const float* pro0 = (const float*)d_in[1];
  const float* pro1 = (const float*)d_in[2];
  const float* di   = (const float*)d_in[3];  // depth_interval
  const int*   cn   = (const int*)d_in[4];    // CostNum
  float* out = (float*)d_out;

  const int N    = in_sizes[0];     // B*H*W = 163840
  const int grid = N / ROWS;        // 640 blocks
  (void)n_in; (void)out_size; (void)d_ws; (void)ws_size;

  hipLaunchKernelGGL(getcost_kernel, dim3(grid), dim3(BDIM), 0, stream,
                     dv, pro0, pro1, di, cn, out);
}